// Quantizer_43233140802034
// MI455X (gfx1250) — compile-verified
//
#include <hip/hip_runtime.h>

typedef __attribute__((ext_vector_type(2))) float v2f;
typedef __attribute__((ext_vector_type(8))) float v8f;

#define B_ 16
#define C_ 128
#define H_ 32
#define W_ 32
#define N_ (B_ * H_ * W_)   /* 16384 */
#define K_ 2048
#define CPAD 132            /* 128 + 4 pad: kills LDS bank conflicts, keeps 16B align */
#define CHUNK 64            /* codebook rows staged in LDS per iteration */

// ---------------------------------------------------------------------------
// Zero the one-hot encodings region (harness poisons d_out with 0xAA).
// ---------------------------------------------------------------------------
__global__ void vq_zero(float4* __restrict__ p, int n4) {
    int i = blockIdx.x * blockDim.x + threadIdx.x;
    int stride = gridDim.x * blockDim.x;
    float4 z; z.x = 0.f; z.y = 0.f; z.z = 0.f; z.w = 0.f;
    for (; i < n4; i += stride) p[i] = z;
}

// ---------------------------------------------------------------------------
// ||w||^2 per codebook row -> workspace.  (x_sq is not needed for argmin.)
// ---------------------------------------------------------------------------
__global__ void vq_wsq(const float* __restrict__ cb, float* __restrict__ wsq) {
    int k = blockIdx.x * blockDim.x + threadIdx.x;
    if (k >= K_) return;
    const float4* row = (const float4*)(cb + (size_t)k * C_);
    float s = 0.f;
    #pragma unroll 8
    for (int i = 0; i < C_ / 4; ++i) {
        float4 v = row[i];
        s += v.x * v.x + v.y * v.y + v.z * v.z + v.w * v.w;
    }
    wsq[k] = s;
}

// ---------------------------------------------------------------------------
// Main kernel: 64 threads (2 wave32s) per block; block owns one (b,h) pair
// = 32 rows of x_flat (w = 0..31).  Wave 0 -> rows 0..15, wave 1 -> 16..31.
// Full fp32 WMMA 16x16x4 over C=128 using TWO independent accumulator
// chains (c-halves) so the 32 WMMAs are not one serial dependency chain.
// ---------------------------------------------------------------------------
__global__ __launch_bounds__(64) void vq_main(const float* __restrict__ x,
                                              const float* __restrict__ cb,
                                              const float* __restrict__ wsq,
                                              float* __restrict__ enc,    // [N,K]
                                              float* __restrict__ quant)  // [B,C,H,W]
{
    __shared__ float lds_x[32][CPAD];        // x tile, transposed: [w][c]
    __shared__ float lds_cb[CHUNK][CPAD];    // codebook chunk: [code][c]
    __shared__ int   lds_idx[32];            // argmin per row

    const int tid   = threadIdx.x;           // 0..63
    const int nbase = blockIdx.x * 32;       // flat NHWC row base
    const int b     = nbase / (H_ * W_);
    const int h     = (nbase % (H_ * W_)) / W_;

    // ---- Load x tile coalesced along w, transpose into LDS -----------------
    // x[((b*C + c)*H + h)*W + w] is contiguous in w (32 floats per c).
    {
        const float4* src = (const float4*)(x + ((size_t)b * C_ * H_ + h) * (size_t)W_);
        #pragma unroll
        for (int i = 0; i < 16; ++i) {
            int idx = tid + i * 64;              // 0..1023 over (c, w4)
            int c   = idx >> 3;                  // 0..127
            int w4  = idx & 7;                   // 0..7
            float4 v = src[(size_t)c * (H_ * W_ / 4) + w4];
            int w = w4 * 4;
            lds_x[w + 0][c] = v.x;
            lds_x[w + 1][c] = v.y;
            lds_x[w + 2][c] = v.z;
            lds_x[w + 3][c] = v.w;
        }
    }

    const int lane = tid & 31;
    const int half = tid >> 5;          // wave id: row group 16*half
    const int ln   = lane & 15;         // A: M index / B: N index
    const int koff = (lane >> 4) * 2;   // K sub-offset per ISA 16x4 layout

    float bestv[8];
    int   besti[8];
    #pragma unroll
    for (int j = 0; j < 8; ++j) { bestv[j] = 3.4e38f; besti[j] = 0; }

    for (int ch = 0; ch < K_ / CHUNK; ++ch) {
        __syncthreads();
        // ---- Stage CHUNK codebook rows into LDS, fully coalesced ----------
        {
            const float4* src = (const float4*)(cb + (size_t)ch * CHUNK * C_);
            #pragma unroll
            for (int i = 0; i < (CHUNK * C_ / 4) / 64; ++i) {   // 32 iters
                int idx  = tid + i * 64;                        // 0..2047
                int code = idx >> 5;
                int c4   = idx & 31;
                *(float4*)&lds_cb[code][c4 * 4] = src[idx];
            }
        }
        __syncthreads();

        for (int kt = 0; kt < CHUNK / 16; ++kt) {
            v8f acc0 = {};                       // c in [0,64)
            v8f acc1 = {};                       // c in [64,128)
            const float* arow = &lds_x[16 * half + ln][koff];
            const float* brow = &lds_cb[kt * 16 + ln][koff];
            #pragma unroll
            for (int cs = 0; cs < C_ / 8; ++cs) {    // 16 iters x 2 wmma
                v2f a0, b0, a1, b1;
                a0.x = arow[cs * 4];            a0.y = arow[cs * 4 + 1];
                b0.x = brow[cs * 4];            b0.y = brow[cs * 4 + 1];
                a1.x = arow[(C_ / 2) + cs * 4]; a1.y = arow[(C_ / 2) + cs * 4 + 1];
                b1.x = brow[(C_ / 2) + cs * 4]; b1.y = brow[(C_ / 2) + cs * 4 + 1];
                acc0 = __builtin_amdgcn_wmma_f32_16x16x4_f32(
                    false, a0, false, b0, (short)0, acc0, false, false);
                acc1 = __builtin_amdgcn_wmma_f32_16x16x4_f32(
                    false, a1, false, b1, (short)0, acc1, false, false);
            }
            // acc[j] = dot(x_row, w_code): lane holds col n = ln,
            // rows j + 8*(lane>>4) of this wave's 16-row tile.
            int   code = ch * CHUNK + kt * 16 + ln;
            float ws   = wsq[code];
            #pragma unroll
            for (int j = 0; j < 8; ++j) {
                float score = ws - 2.0f * (acc0[j] + acc1[j]);  // = d^2 - ||x||^2
                if (score < bestv[j]) { bestv[j] = score; besti[j] = code; }
            }
        }
    }

    // ---- argmin reduce over the 16 lanes of each half-wave ------------------
    #pragma unroll
    for (int j = 0; j < 8; ++j) {
        float v  = bestv[j];
        int   ix = besti[j];
        #pragma unroll
        for (int off = 1; off < 16; off <<= 1) {
            float ov = __shfl_xor(v, off, 32);
            int   oi = __shfl_xor(ix, off, 32);
            if (ov < v || (ov == v && oi < ix)) { v = ov; ix = oi; }  // first-min tie-break
        }
        if (ln == 0) {
            int row = 16 * half + 8 * (lane >> 4) + j;   // row within 32-row tile
            lds_idx[row] = ix;
        }
    }
    __syncthreads();

    // ---- one-hot scatter (region pre-zeroed) --------------------------------
    if (tid < 32) {
        int n = nbase + tid;
        enc[(size_t)n * K_ + lds_idx[tid]] = 1.0f;
    }

    // ---- quantized gather-write, coalesced along w --------------------------
    #pragma unroll 4
    for (int i = 0; i < 64; ++i) {
        int idx = tid + i * 64;          // 0..4095 over (c, w)
        int c   = idx >> 5;
        int w   = idx & 31;
        quant[((size_t)(b * C_ + c) * H_ + h) * W_ + w] =
            cb[(size_t)lds_idx[w] * C_ + c];
    }
}

// ---------------------------------------------------------------------------
extern "C" void kernel_launch(void* const* d_in, const int* in_sizes, int n_in,
                              void* d_out, int out_size, void* d_ws, size_t ws_size,
                              hipStream_t stream) {
    const float* x  = (const float*)d_in[0];   // [16,128,32,32] f32
    const float* cb = (const float*)d_in[1];   // [2048,128] f32

    float* enc   = (float*)d_out;              // [N,K] one-hot
    float* quant = enc + (size_t)N_ * K_;      // [B,C,H,W]
    float* wsq   = (float*)d_ws;               // 2048 floats scratch

    int n4 = N_ * K_ / 4;                      // 8,388,608 float4
    vq_zero<<<4096, 256, 0, stream>>>((float4*)enc, n4);
    vq_wsq<<<K_ / 256, 256, 0, stream>>>(cb, wsq);
    vq_main<<<N_ / 32, 64, 0, stream>>>(x, cb, wsq, enc, quant);
}